// XSimGCL_Encoder_27041114095964
// MI455X (gfx1250) — compile-verified
//
#include <hip/hip_runtime.h>

#define USER_NUM  100000
#define ITEM_NUM  50000
#define N_NODES   150000
#define EMB       64
#define NNZ       6400000
#define N_LAYERS  3
#define LEAKY     0.01f
#define NORM_EPS  1e-12f

typedef __attribute__((ext_vector_type(2))) float v2f;
typedef __attribute__((ext_vector_type(8))) float v8f;

__device__ __forceinline__ unsigned mix32(unsigned x) {
    x ^= x >> 16; x *= 0x7feb352dU;
    x ^= x >> 15; x *= 0x846ca68bU;
    x ^= x >> 16;
    return x;
}

__device__ __forceinline__ void atomAddF(float* p, float v) {
    __hip_atomic_fetch_add(p, v, __ATOMIC_RELAXED, __HIP_MEMORY_SCOPE_AGENT);
}

// ---------------------------------------------------------------------------
// Fused 3-channel SpMM: one pass over the edge list feeds ego/img/txt.
// One edge per wave inner-iteration: 32 lanes x float2 = 64-float row.
// Gathers are global_load_b64; scatters are global_atomic_add_f32 (L2-side).
// ---------------------------------------------------------------------------
__global__ void spmm3_kernel(const float* __restrict__ egoIn,
                             const float* __restrict__ imgIn,
                             const float* __restrict__ txtIn,
                             float* __restrict__ egoOut,
                             float* __restrict__ imgOut,
                             float* __restrict__ txtOut,
                             const float* __restrict__ avals,
                             const int*   __restrict__ arows,
                             const int*   __restrict__ acols)
{
    const int lane  = threadIdx.x & 31;
    const int wave  = blockIdx.x * (blockDim.x >> 5) + (threadIdx.x >> 5);
    const int nwave = gridDim.x * (blockDim.x >> 5);
    const int off   = lane * 2;

    // NNZ is an exact multiple of 32, so no tail predication is needed.
    for (int base = wave * 32; base < NNZ; base += nwave * 32) {
        const int e = base + lane;
        float v = avals[e];
        int   r = arows[e];
        int   c = acols[e];

        const int nb = base + nwave * 32;
        if (nb < NNZ) {   // gfx1250 speculative prefetch of the edge stream
            __builtin_prefetch(&avals[nb + lane], 0, 0);
            __builtin_prefetch(&arows[nb + lane], 0, 0);
            __builtin_prefetch(&acols[nb + lane], 0, 0);
        }

        #pragma unroll 4
        for (int i = 0; i < 32; ++i) {
            const float ve = __shfl(v, i, 32);
            const int   re = __shfl(r, i, 32);
            const int   ce = __shfl(c, i, 32);
            const int   cb = ce * EMB + off;
            const int   rb = re * EMB + off;

            const float2 eg = *(const float2*)(egoIn + cb);
            const float2 im = *(const float2*)(imgIn + cb);
            const float2 tx = *(const float2*)(txtIn + cb);

            atomAddF(egoOut + rb,     ve * eg.x);
            atomAddF(egoOut + rb + 1, ve * eg.y);
            atomAddF(imgOut + rb,     ve * im.x);
            atomAddF(imgOut + rb + 1, ve * im.y);
            atomAddF(txtOut + rb,     ve * tx.x);
            atomAddF(txtOut + rb + 1, ve * tx.y);
        }
    }
}

// ---------------------------------------------------------------------------
// acc += x (elementwise, float4)
// ---------------------------------------------------------------------------
__global__ void acc_add_kernel(float* __restrict__ acc,
                               const float* __restrict__ x, int n4)
{
    int i = blockIdx.x * blockDim.x + threadIdx.x;
    const int stride = gridDim.x * blockDim.x;
    for (; i < n4; i += stride) {
        float4 a = ((float4*)acc)[i];
        const float4 b = ((const float4*)x)[i];
        a.x += b.x; a.y += b.y; a.z += b.z; a.w += b.w;
        ((float4*)acc)[i] = a;
    }
}

// ---------------------------------------------------------------------------
// img/txt post-pass: leaky-relu -> dropout(0.5) -> write back (next layer
// input) -> per-row L2 norm via V_WMMA_F32_16X16X4_F32 against an all-ones
// B matrix (D[m][n] = sum_k A[m][k], i.e. row sum-of-squares replicated
// across columns) -> acc += normalized.
// One wave owns a 16-row x 64-col tile. N_NODES = 16 * 9375 exactly.
// A-matrix f32 16x4 layout: lanes 0-15 hold K=0,1; lanes 16-31 hold K=2,3.
// ---------------------------------------------------------------------------
__global__ void post_norm_kernel(float* __restrict__ buf,
                                 float* __restrict__ acc,
                                 unsigned seed)
{
    const int lane  = threadIdx.x & 31;
    const int m     = lane & 15;       // row within 16-row tile
    const int kh    = lane >> 4;       // K half: 0 -> K{0,1}, 1 -> K{2,3}
    const int wave  = blockIdx.x * (blockDim.x >> 5) + (threadIdx.x >> 5);
    const int nwave = gridDim.x * (blockDim.x >> 5);
    const int ntiles = N_NODES / 16;

    v2f ones; ones[0] = 1.0f; ones[1] = 1.0f;

    for (int tile = wave; tile < ntiles; tile += nwave) {
        const int row = tile * 16 + m;
        float* rbase = buf + row * EMB;

        v8f cacc = {};
        float2 vals[16];

        #pragma unroll
        for (int k = 0; k < 16; ++k) {
            const int col = 4 * k + 2 * kh;
            float2 f = *(const float2*)(rbase + col);
            // leaky relu
            f.x = f.x > 0.0f ? f.x : LEAKY * f.x;
            f.y = f.y > 0.0f ? f.y : LEAKY * f.y;
            // dropout p=0.5, scale 2.0 (deterministic counter-based hash)
            const unsigned idx = (unsigned)row * EMB + (unsigned)col;
            f.x = (mix32(idx ^ seed)       >> 31) ? f.x * 2.0f : 0.0f;
            f.y = (mix32((idx + 1) ^ seed) >> 31) ? f.y * 2.0f : 0.0f;
            vals[k] = f;
            *(float2*)(rbase + col) = f;     // feeds next layer's SpMM

            v2f a; a[0] = f.x * f.x; a[1] = f.y * f.y;
            // D = A(16x4 squared) x ones(4x16) + C  -> row sums-of-squares
            cacc = __builtin_amdgcn_wmma_f32_16x16x4_f32(
                false, a, false, ones, (short)0, cacc, false, false);
        }

        // D vgpr r: lanes 0-15 hold M=r, lanes 16-31 hold M=8+r.
        float rs = 0.0f;
        #pragma unroll
        for (int r = 0; r < 8; ++r) {
            const float lo = __shfl(cacc[r], 0, 32);   // rowsum[r]
            const float hi = __shfl(cacc[r], 16, 32);  // rowsum[8+r]
            rs = (m == r)     ? lo : rs;
            rs = (m == r + 8) ? hi : rs;
        }
        const float inv = 1.0f / fmaxf(sqrtf(rs), NORM_EPS);

        float* abase = acc + row * EMB;
        #pragma unroll
        for (int k = 0; k < 16; ++k) {
            const int col = 4 * k + 2 * kh;
            float2 a2 = *(float2*)(abase + col);
            a2.x += vals[k].x * inv;
            a2.y += vals[k].y * inv;
            *(float2*)(abase + col) = a2;
        }
    }
}

// ---------------------------------------------------------------------------
// out[:U*64]  = egoAcc/3 ; out[U*64:] = (w0*egoAcc + w1*imgAcc + w2*txtAcc)/3
// ---------------------------------------------------------------------------
__global__ void finalize_kernel(const float* __restrict__ egoAcc,
                                const float* __restrict__ imgAcc,
                                const float* __restrict__ txtAcc,
                                const float* __restrict__ w,
                                float* __restrict__ out)
{
    const float inv_l = 1.0f / (float)N_LAYERS;
    const float w0 = w[0], w1 = w[1], w2 = w[2];
    const int userElems = USER_NUM * EMB;
    const int totElems  = N_NODES * EMB;
    int i = blockIdx.x * blockDim.x + threadIdx.x;
    const int stride = gridDim.x * blockDim.x;
    for (; i < totElems; i += stride) {
        if (i < userElems) {
            out[i] = egoAcc[i] * inv_l;
        } else {
            out[i] = (w0 * egoAcc[i] + w1 * imgAcc[i] + w2 * txtAcc[i]) * inv_l;
        }
    }
}

// ---------------------------------------------------------------------------
extern "C" void kernel_launch(void* const* d_in, const int* in_sizes, int n_in,
                              void* d_out, int out_size, void* d_ws, size_t ws_size,
                              hipStream_t stream)
{
    (void)in_sizes; (void)n_in; (void)out_size; (void)ws_size;

    const float* user  = (const float*)d_in[0];
    const float* item  = (const float*)d_in[1];
    const float* image = (const float*)d_in[2];
    const float* text  = (const float*)d_in[3];
    const float* fw    = (const float*)d_in[4];
    const float* avals = (const float*)d_in[5];
    const int*   arows = (const int*)d_in[6];
    const int*   acols = (const int*)d_in[7];

    const size_t BUF = (size_t)N_NODES * EMB;          // floats per buffer
    float* ws = (float*)d_ws;
    float* ego[2] = { ws + 0 * BUF, ws + 1 * BUF };
    float* img[2] = { ws + 2 * BUF, ws + 3 * BUF };
    float* txt[2] = { ws + 4 * BUF, ws + 5 * BUF };
    float* egoAcc = ws + 6 * BUF;
    float* imgAcc = ws + 7 * BUF;
    float* txtAcc = ws + 8 * BUF;

    const size_t UB = (size_t)USER_NUM * EMB * sizeof(float);
    const size_t IB = (size_t)ITEM_NUM * EMB * sizeof(float);
    const size_t UO = (size_t)USER_NUM * EMB;

    // ego0 = [user; item], img0 = [user; image], txt0 = [user; text]
    hipMemcpyAsync(ego[0],      user,  UB, hipMemcpyDeviceToDevice, stream);
    hipMemcpyAsync(ego[0] + UO, item,  IB, hipMemcpyDeviceToDevice, stream);
    hipMemcpyAsync(img[0],      user,  UB, hipMemcpyDeviceToDevice, stream);
    hipMemcpyAsync(img[0] + UO, image, IB, hipMemcpyDeviceToDevice, stream);
    hipMemcpyAsync(txt[0],      user,  UB, hipMemcpyDeviceToDevice, stream);
    hipMemcpyAsync(txt[0] + UO, text,  IB, hipMemcpyDeviceToDevice, stream);

    // accumulators (contiguous) -> zero
    hipMemsetAsync(egoAcc, 0, 3 * BUF * sizeof(float), stream);

    int cur = 0;
    for (int layer = 0; layer < N_LAYERS; ++layer) {
        const int nxt = cur ^ 1;
        hipMemsetAsync(ego[nxt], 0, BUF * sizeof(float), stream);
        hipMemsetAsync(img[nxt], 0, BUF * sizeof(float), stream);
        hipMemsetAsync(txt[nxt], 0, BUF * sizeof(float), stream);

        spmm3_kernel<<<4096, 256, 0, stream>>>(
            ego[cur], img[cur], txt[cur],
            ego[nxt], img[nxt], txt[nxt],
            avals, arows, acols);

        acc_add_kernel<<<1024, 256, 0, stream>>>(egoAcc, ego[nxt], (int)(BUF / 4));

        const unsigned s0 = 0x9E3779B1u * (unsigned)(layer * 2 + 1);
        const unsigned s1 = 0x9E3779B1u * (unsigned)(layer * 2 + 2);
        post_norm_kernel<<<1172, 256, 0, stream>>>(img[nxt], imgAcc, s0);
        post_norm_kernel<<<1172, 256, 0, stream>>>(txt[nxt], txtAcc, s1);

        cur = nxt;
    }

    finalize_kernel<<<2048, 256, 0, stream>>>(egoAcc, imgAcc, txtAcc, fw,
                                              (float*)d_out);
}